// SAGENet_35150012351301
// MI455X (gfx1250) — compile-verified
//
#include <hip/hip_runtime.h>
#include <hip/hip_bf16.h>
#include <cstddef>

typedef __attribute__((ext_vector_type(2))) float v2f;
typedef __attribute__((ext_vector_type(8))) float v8f;

// ---------------------------------------------------------------------------
// Zero the scratch region (deg + buf1 + buf2) each call for determinism.
// ---------------------------------------------------------------------------
__global__ void zero_ws_kernel(float* __restrict__ p, int n) {
    int i = blockIdx.x * blockDim.x + threadIdx.x;
    if (i < n) p[i] = 0.0f;
}

// ---------------------------------------------------------------------------
// Edge scatter-sum: 16 threads per edge, each handles 4 contiguous channels
// via a float4 gather + 4 global f32 atomic adds. Thread 0 of each edge
// optionally accumulates the in-degree.
// ---------------------------------------------------------------------------
__global__ void scatter_sum_kernel(const float* __restrict__ feat,
                                   const int* __restrict__ src,
                                   const int* __restrict__ dst,
                                   float* __restrict__ acc,
                                   float* __restrict__ deg,
                                   int E, int accum_deg) {
    int tid = blockIdx.x * blockDim.x + threadIdx.x;
    int e = tid >> 4;
    if (e >= E) return;
    int c = (tid & 15) << 2;                       // channel group base (0..60)
    int s = src[e];
    int d = dst[e];
    const float4 v = *reinterpret_cast<const float4*>(feat + (size_t)s * 64 + c);
    float* o = acc + (size_t)d * 64 + c;
    atomicAdd(o + 0, v.x);
    atomicAdd(o + 1, v.y);
    atomicAdd(o + 2, v.z);
    atomicAdd(o + 3, v.w);
    if (accum_deg && (tid & 15) == 0) atomicAdd(deg + d, 1.0f);
}

// ---------------------------------------------------------------------------
// Layer 1: h = relu( (agg/deg) @ W1 + b1 ), 64->64, via V_WMMA_F32_16X16X4_F32.
// One wave per 16-node tile. A fragment layout (16x4 f32):
//   lanes 0-15: M=lane, holds K = 4k+0 / 4k+1 ; lanes 16-31: M=lane-16, K=4k+2/4k+3
// C/D layout: VGPR r -> row (r + 8*half), column = lane&15.
// In-place output is safe (row-block independent).
// ---------------------------------------------------------------------------
__global__ void sage_layer1_wmma(const float* __restrict__ aggin,
                                 const float* __restrict__ deg,
                                 const float* __restrict__ W1,
                                 const float* __restrict__ b1,
                                 float* __restrict__ hout, int n) {
    const int wave = (blockIdx.x * blockDim.x + threadIdx.x) >> 5;
    const int lane = threadIdx.x & 31;
    const int row0 = wave * 16;
    if (row0 >= n) return;                          // wave-uniform exit
    __builtin_prefetch(W1, 0, 3);                   // global_prefetch_b8: W1 is hot
    const int m    = lane & 15;
    const int half = lane >> 4;
    const int koff = half * 2;
    int row = row0 + m;
    if (row >= n) row = n - 1;
    const float invd = 1.0f / fmaxf(deg[row], 1.0f);

    // Build all 16 A fragments once (reused for all 4 output column tiles).
    v2f a[16];
    const float* arow = aggin + (size_t)row * 64 + koff;
#pragma unroll
    for (int k = 0; k < 16; ++k) {
        v2f t = *reinterpret_cast<const v2f*>(arow + 4 * k);
        a[k].x = t.x * invd;
        a[k].y = t.y * invd;
    }

#pragma unroll
    for (int nt = 0; nt < 4; ++nt) {
        const int col = nt * 16 + m;
        v8f acc = {};
#pragma unroll
        for (int k = 0; k < 16; ++k) {
            v2f b;                                  // B fragment: rows 4k..4k+3 of W1
            b.x = W1[(4 * k + koff) * 64 + col];
            b.y = W1[(4 * k + koff + 1) * 64 + col];
            acc = __builtin_amdgcn_wmma_f32_16x16x4_f32(
                false, a[k], false, b, (short)0, acc, false, false);
        }
        const float bias = b1[col];
#pragma unroll
        for (int r = 0; r < 8; ++r) {
            const int orow = row0 + half * 8 + r;
            if (orow < n)
                hout[(size_t)orow * 64 + col] = fmaxf(acc[r] + bias, 0.0f);
        }
    }
}

// ---------------------------------------------------------------------------
// Layer 2: out = log_softmax( (agg2/deg) @ W2 + b2 ), 64->16.
// Single 16-wide column tile. Each output row's 16 logits live in one VGPR
// across a 16-lane half-wave -> softmax reductions are 4 shfl_xor steps.
// ---------------------------------------------------------------------------
__global__ void sage_layer2_wmma(const float* __restrict__ aggin,
                                 const float* __restrict__ deg,
                                 const float* __restrict__ W2,
                                 const float* __restrict__ b2,
                                 float* __restrict__ out, int n) {
    const int wave = (blockIdx.x * blockDim.x + threadIdx.x) >> 5;
    const int lane = threadIdx.x & 31;
    const int row0 = wave * 16;
    if (row0 >= n) return;
    __builtin_prefetch(W2, 0, 3);
    const int m    = lane & 15;
    const int half = lane >> 4;
    const int koff = half * 2;
    int row = row0 + m;
    if (row >= n) row = n - 1;
    const float invd = 1.0f / fmaxf(deg[row], 1.0f);

    v8f acc = {};
    const float* arow = aggin + (size_t)row * 64 + koff;
#pragma unroll
    for (int k = 0; k < 16; ++k) {
        v2f t = *reinterpret_cast<const v2f*>(arow + 4 * k);
        v2f a;
        a.x = t.x * invd;
        a.y = t.y * invd;
        v2f b;                                      // W2 is 64x16 row-major
        b.x = W2[(4 * k + koff) * 16 + m];
        b.y = W2[(4 * k + koff + 1) * 16 + m];
        acc = __builtin_amdgcn_wmma_f32_16x16x4_f32(
            false, a, false, b, (short)0, acc, false, false);
    }
    const float bias = b2[m];
#pragma unroll
    for (int r = 0; r < 8; ++r) {
        float z = acc[r] + bias;
        float mx = z;                                // row max across 16 lanes
        mx = fmaxf(mx, __shfl_xor(mx, 1, 32));
        mx = fmaxf(mx, __shfl_xor(mx, 2, 32));
        mx = fmaxf(mx, __shfl_xor(mx, 4, 32));
        mx = fmaxf(mx, __shfl_xor(mx, 8, 32));
        float s = __expf(z - mx);                    // row sum of exp
        s += __shfl_xor(s, 1, 32);
        s += __shfl_xor(s, 2, 32);
        s += __shfl_xor(s, 4, 32);
        s += __shfl_xor(s, 8, 32);
        const int orow = row0 + half * 8 + r;
        if (orow < n)
            out[(size_t)orow * 16 + m] = z - mx - __logf(s);
    }
}

// ---------------------------------------------------------------------------
// Launch: zero ws -> scatter1(+deg) -> layer1 WMMA (in place) ->
//         scatter2 -> layer2 WMMA + log_softmax -> d_out
// ---------------------------------------------------------------------------
extern "C" void kernel_launch(void* const* d_in, const int* in_sizes, int n_in,
                              void* d_out, int out_size, void* d_ws, size_t ws_size,
                              hipStream_t stream) {
    const float* x  = (const float*)d_in[0];
    const int*  ei  = (const int*)d_in[1];
    const float* W1 = (const float*)d_in[2];
    const float* b1 = (const float*)d_in[3];
    const float* W2 = (const float*)d_in[4];
    const float* b2 = (const float*)d_in[5];
    float* out = (float*)d_out;

    const int N = in_sizes[0] / 64;
    const int E = in_sizes[1] / 2;
    const int* src = ei;
    const int* dst = ei + E;

    float* deg  = (float*)d_ws;                 // N floats
    float* buf1 = deg + N;                      // N*64 floats (agg1 -> h in place)
    float* buf2 = buf1 + (size_t)N * 64;        // N*64 floats (agg2)

    {   // zero deg|buf1|buf2 (contiguous N*129 floats)
        int total  = N * 129;
        int blocks = (total + 255) / 256;
        zero_ws_kernel<<<blocks, 256, 0, stream>>>((float*)d_ws, total);
    }
    {   // scatter x -> buf1, accumulate degree
        int total  = E * 16;
        int blocks = (total + 255) / 256;
        scatter_sum_kernel<<<blocks, 256, 0, stream>>>(x, src, dst, buf1, deg, E, 1);
    }
    {   // layer 1: mean + GEMM(64x64) + bias + relu, in place
        int waves  = (N + 15) / 16;
        int blocks = (waves + 7) / 8;           // 8 waves (256 threads) per block
        sage_layer1_wmma<<<blocks, 256, 0, stream>>>(buf1, deg, W1, b1, buf1, N);
    }
    {   // scatter h -> buf2
        int total  = E * 16;
        int blocks = (total + 255) / 256;
        scatter_sum_kernel<<<blocks, 256, 0, stream>>>(buf1, src, dst, buf2, deg, E, 0);
    }
    {   // layer 2: mean + GEMM(64x16) + bias + log_softmax
        int waves  = (N + 15) / 16;
        int blocks = (waves + 7) / 8;
        sage_layer2_wmma<<<blocks, 256, 0, stream>>>(buf2, deg, W2, b2, out, N);
    }
}